// LovaszSoftmaxV3_14156212208199
// MI455X (gfx1250) — compile-verified
//
#include <hip/hip_runtime.h>
#include <hip/hip_bf16.h>
#include <stdint.h>

// ----------------------------------------------------------------------------
// Lovasz-Softmax loss, MI455X (gfx1250).
//  - No matmul in this op -> WMMA not applicable; the CDNA5 paths that matter
//    are TDM (tensor_load_to_lds / TENSORcnt), LDS, wave32, u64 global atomics.
//  - Sort is replaced by an exact-within-tie-groups counting sort on the
//    float bit pattern of the per-class error (monotone for e in [0,1]).
//  - hist[c][key] packs {pos_count:32, total_count:32} per bin, built with one
//    global_atomic_add_u64 per (pixel, class).
//  - Per-class descending scan over bins reconstructs cumulative counts and
//    accumulates  v_hat * (J(cum) - J(cum - own))  with J the Lovasz jaccard.
// ----------------------------------------------------------------------------

#define LV_C     19
#define LV_HW    262144u   // 512*512
#define LV_TILE  256
#define LV_IGN   (-100)

typedef unsigned int lv_u32x4 __attribute__((ext_vector_type(4)));
typedef int          lv_i32x4 __attribute__((ext_vector_type(4)));
typedef int          lv_i32x8 __attribute__((ext_vector_type(8)));

// ---------------------------------------------------------------- zero ws ---
__global__ __launch_bounds__(256) void lv_zero_kernel(unsigned long long* __restrict__ p,
                                                      size_t n) {
  size_t i = (size_t)blockIdx.x * blockDim.x + threadIdx.x;
  const size_t stride = (size_t)gridDim.x * blockDim.x;
  for (; i < n; i += stride) p[i] = 0ull;
}

// ------------------------------------------------- softmax + histogram ------
__global__ __launch_bounds__(256) void lv_hist_kernel(const float* __restrict__ logits,
                                                      const int* __restrict__ labels,
                                                      unsigned long long* __restrict__ hist,
                                                      unsigned int NB, unsigned int shift) {
  __shared__ float tile[LV_C * LV_TILE];  // 19 rows x 256 pixels = 19 KB

  const int tid = threadIdx.x;
  const unsigned int p0  = (unsigned int)blockIdx.x << 8;  // 256 pixels / block
  const unsigned int n   = p0 / LV_HW;                     // batch index
  const unsigned int hw0 = p0 % LV_HW;                     // offset in H*W
  const float* gbase = logits + (size_t)n * LV_C * LV_HW + hw0;

  // Wave 0 issues one TDM 2D tile load: 19 rows x 1KB, row stride = HW floats.
  if (tid < 32) {
    const unsigned long long ga = (unsigned long long)(uintptr_t)gbase;
    const unsigned int ldsb = (unsigned int)(uintptr_t)&tile[0];

    lv_u32x4 g0;
    g0[0] = 1u;                                              // count=1 valid D#
    g0[1] = ldsb;                                            // lds_addr
    g0[2] = (unsigned int)ga;                                // global_addr[31:0]
    g0[3] = ((unsigned int)(ga >> 32) & 0x01FFFFFFu)         // global_addr[56:32]
            | 0x80000000u;                                   // type=2 ("image")

    lv_i32x8 g1;
    g1[0] = 0x00020000;                                      // data_size=4B, no mask
    g1[1] = (int)((LV_HW & 0xFFFFu) << 16);                  // tensor_dim0[15:0]
    g1[2] = (int)((LV_HW >> 16) | ((unsigned)LV_C << 16));   // td0[31:16] | td1[15:0]
    g1[3] = (int)((unsigned)LV_TILE << 16);                  // tile_dim0=256
    g1[4] = LV_C;                                            // tile_dim1=19
    g1[5] = (int)LV_HW;                                      // dim0 stride lo
    g1[6] = 0;                                               // stride hi / d1 stride
    g1[7] = 0;

#if defined(__HIP_DEVICE_COMPILE__) && __has_builtin(__builtin_amdgcn_tensor_load_to_lds)
#  if __has_include(<hip/amd_detail/amd_gfx1250_TDM.h>)
    {   // therock / clang-23 6-arg form
      lv_i32x4 z4 = {0, 0, 0, 0};
      lv_i32x8 z8 = {0, 0, 0, 0, 0, 0, 0, 0};
      __builtin_amdgcn_tensor_load_to_lds(g0, g1, z4, z4, z8, 0);
    }
#  else
    {   // ROCm 7.2 / clang-22 5-arg form
      lv_i32x4 z4 = {0, 0, 0, 0};
      __builtin_amdgcn_tensor_load_to_lds(g0, g1, z4, z4, 0);
    }
#  endif
#else
    asm volatile("tensor_load_to_lds %0, %1" :: "s"(g0), "s"(g1) : "memory");
#endif

#if defined(__HIP_DEVICE_COMPILE__) && __has_builtin(__builtin_amdgcn_s_wait_tensorcnt)
    __builtin_amdgcn_s_wait_tensorcnt(0);
#else
    asm volatile("s_wait_tensorcnt 0" ::: "memory");
#endif
  }
  __syncthreads();

  const unsigned int p = p0 + (unsigned int)tid;
  const int lab = labels[p];
  const bool valid = (lab != LV_IGN);

  // softmax over 19 classes; LDS reads are bank-conflict-free (stride 256).
  float x[LV_C];
  float m = -3.402823466e38f;
#pragma unroll
  for (int c = 0; c < LV_C; ++c) {
    x[c] = tile[c * LV_TILE + tid];
    m = fmaxf(m, x[c]);
  }
  float s = 0.f;
#pragma unroll
  for (int c = 0; c < LV_C; ++c) {
    const float e = __expf(x[c] - m);
    x[c] = e;
    s += e;
  }
  const float inv = 1.0f / s;

  if (valid) {
#pragma unroll
    for (int c = 0; c < LV_C; ++c) {
      const float prob = x[c] * inv;
      float err = (c == lab) ? (1.0f - prob) : prob;
      err = fminf(fmaxf(err, 0.0f), 1.0f);
      const unsigned int key = __float_as_uint(err) >> shift;
      const unsigned long long pack = (c == lab) ? 0x100000001ull : 1ull;
      atomicAdd(&hist[(size_t)c * NB + key], pack);  // global_atomic_add_u64
    }
  }
}

// --------------------------------- per-class descending scan over the bins --
__global__ __launch_bounds__(256) void lv_scan_kernel(const unsigned long long* __restrict__ hist,
                                                      float* __restrict__ losses,
                                                      unsigned int NB, unsigned int shift) {
  const int c = blockIdx.x;
  const int tid = threadIdx.x;
  const unsigned long long* h = hist + (size_t)c * NB;

  __shared__ unsigned long long sbuf[256];
  __shared__ unsigned long long carry_sh;
  __shared__ float fred[256];

  // Pass 1: n_pos = sum of high words (packed sums can't carry: counts < 2^32).
  unsigned long long acc = 0ull;
  for (unsigned int i = tid; i < NB; i += 256u) acc += h[i];
  sbuf[tid] = acc;
  __syncthreads();
  for (int off = 128; off > 0; off >>= 1) {
    if (tid < off) sbuf[tid] += sbuf[tid + off];
    __syncthreads();
  }
  const float n_pos = (float)(unsigned int)(sbuf[0] >> 32);
  __syncthreads();

  // Pass 2: descending cumulative scan; tid ascending == key descending.
  unsigned long long carry = 0ull;
  float loss = 0.f;
  const int nch = (int)(NB >> 8);
  for (int cb = nch - 1; cb >= 0; --cb) {
    const unsigned int k = (unsigned int)cb * 256u + (255u - (unsigned int)tid);
    const unsigned long long v = h[k];

    sbuf[tid] = v;
    __syncthreads();
    for (int off = 1; off < 256; off <<= 1) {   // inclusive Hillis-Steele
      const unsigned long long t = (tid >= off) ? sbuf[tid - off] : 0ull;
      __syncthreads();
      sbuf[tid] += t;
      __syncthreads();
    }
    const unsigned long long cum = carry + sbuf[tid];

    if (v != 0ull) {
      const unsigned int ct = (unsigned int)cum;          // cumulative total
      const unsigned int cp = (unsigned int)(cum >> 32);  // cumulative positives
      const unsigned int nt = (unsigned int)v;
      const unsigned int np = (unsigned int)(v >> 32);
      const float J = 1.f - (n_pos - (float)cp) / (n_pos + (float)(ct - cp));
      const unsigned int pt = ct - nt, pp = cp - np;
      const float Jp = (pt == 0u)
                         ? 0.f
                         : 1.f - (n_pos - (float)pp) / (n_pos + (float)(pt - pp));
      const unsigned int vb = (k << shift) | (1u << (shift - 1));  // bin midpoint
      loss += __uint_as_float(vb) * (J - Jp);
    }
    if (tid == 255) carry_sh = cum;
    __syncthreads();
    carry = carry_sh;   // next loop's first barrier orders this read vs. rewrite
  }

  fred[tid] = loss;
  __syncthreads();
  for (int off = 128; off > 0; off >>= 1) {
    if (tid < off) fred[tid] += fred[tid + off];
    __syncthreads();
  }
  if (tid == 0) losses[c] = fred[0];
}

// -------------------------------------------------------- final mean --------
__global__ void lv_final_kernel(const float* __restrict__ losses, float* __restrict__ out) {
  if (threadIdx.x == 0 && blockIdx.x == 0) {
    float s = 0.f;
    for (int c = 0; c < LV_C; ++c) s += losses[c];
    out[0] = s / (float)LV_C;
  }
}

// ----------------------------------------------------------------------------
extern "C" void kernel_launch(void* const* d_in, const int* in_sizes, int n_in,
                              void* d_out, int out_size, void* d_ws, size_t ws_size,
                              hipStream_t stream) {
  const float* logits = (const float*)d_in[0];
  const int* labels = (const int*)d_in[1];
  float* out = (float*)d_out;

  const unsigned int P = (unsigned int)in_sizes[1];  // 8*512*512 = 2,097,152

  // Pick the finest key quantization whose histogram fits the workspace.
  size_t avail = (ws_size > 4096) ? ws_size - 4096 : 0;
  unsigned int shift = 12, NB = 0;
  for (; shift <= 23u; ++shift) {
    const unsigned int nb = (((0x3F800000u >> shift) + 1u) + 255u) & ~255u;
    if ((size_t)LV_C * nb * 8ull <= avail || shift == 23u) { NB = nb; break; }
  }

  unsigned long long* hist = (unsigned long long*)d_ws;
  float* losses = (float*)((char*)d_ws + (size_t)LV_C * NB * 8ull);
  const size_t nwords = (size_t)LV_C * NB;

  lv_zero_kernel<<<512, 256, 0, stream>>>(hist, nwords);
  lv_hist_kernel<<<P / LV_TILE, 256, 0, stream>>>(logits, labels, hist, NB, shift);
  lv_scan_kernel<<<LV_C, 256, 0, stream>>>(hist, losses, NB, shift);
  lv_final_kernel<<<1, 32, 0, stream>>>(losses, out);
}